// Diagonal_80281528696879
// MI455X (gfx1250) — compile-verified
//
#include <hip/hip_runtime.h>
#include <hip/hip_bf16.h>

// out[r,c] = x[r,c] * fabs(W[c,c])   (B=65536, D=1024, fp32)
// Memory-bound: 512MB traffic -> ~22us floor at 23.3 TB/s. No WMMA (would add
// 16x FLOPs through the slow f32 matrix path with zero bandwidth benefit).
// CDNA5-specific paths used: async global->LDS staging of the diagonal
// (ASYNCcnt + s_wait_asynccnt), b128 streaming with non-temporal hints.

typedef float v4f __attribute__((ext_vector_type(4)));

#define D_DIM 1024
#define ROWS_PER_BLOCK 32

// ---------------------------------------------------------------------------
// Kernel 1: extract |diag(W)| (strided gather, 4KB total) into workspace.
// ---------------------------------------------------------------------------
__global__ __launch_bounds__(256) void diag_abs_kernel(
    const float* __restrict__ W, float* __restrict__ dabs) {
  int c = blockIdx.x * 256 + threadIdx.x;
  if (c < D_DIM) {
    dabs[c] = fabsf(W[(size_t)c * (D_DIM + 1)]);
  }
}

// ---------------------------------------------------------------------------
// Kernel 2: streaming scale. 256 threads = D/4 float4 columns, one per thread.
// Diagonal staged into LDS via gfx1250 async global->LDS b128 copy.
// ---------------------------------------------------------------------------
__global__ __launch_bounds__(256) void scale_kernel(
    const float* __restrict__ x, const float* __restrict__ dabs,
    float* __restrict__ out, int nrows) {
  __shared__ float sdiag[D_DIM];

  const unsigned tid = threadIdx.x;

  // --- Stage 4KB diagonal into LDS with the CDNA5 async-copy path. ---
  // Each of the 256 lanes moves 16 bytes: global(dabs + 16*tid) -> LDS(16*tid).
  {
    unsigned lds_off = (unsigned)(unsigned long long)(&sdiag[0]) + tid * 16u;
    unsigned g_off = tid * 16u;
    asm volatile(
        "global_load_async_to_lds_b128 %0, %1, %2"
        :
        : "v"(lds_off), "v"(g_off), "s"(dabs)
        : "memory");
    asm volatile("s_wait_asynccnt 0" ::: "memory");
  }
  __syncthreads();

  // Per-thread diagonal vector lives in registers for the whole row loop.
  const v4f dv = ((const v4f*)sdiag)[tid];

  const v4f* __restrict__ xv = (const v4f*)x;
  v4f* __restrict__ ov = (v4f*)out;

  const int row0 = blockIdx.x * ROWS_PER_BLOCK;

#pragma unroll 8
  for (int r = 0; r < ROWS_PER_BLOCK; ++r) {
    int row = row0 + r;
    if (row < nrows) {
      size_t idx = (size_t)row * (D_DIM / 4) + tid;
      // Pure stream, no reuse: non-temporal load+store keeps L2 clean.
      v4f v = __builtin_nontemporal_load(&xv[idx]);
      v *= dv;
      __builtin_nontemporal_store(v, &ov[idx]);
    }
  }
}

extern "C" void kernel_launch(void* const* d_in, const int* in_sizes, int n_in,
                              void* d_out, int out_size, void* d_ws, size_t ws_size,
                              hipStream_t stream) {
  const float* x = (const float*)d_in[0];   // [B, D] fp32
  const float* W = (const float*)d_in[1];   // [D, D] fp32
  float* out = (float*)d_out;               // [B, D] fp32
  float* dabs = (float*)d_ws;               // 4KB scratch for |diag(W)|

  const int B = in_sizes[0] / D_DIM;

  // 1) |diag(W)| -> workspace (contiguous 4KB).
  diag_abs_kernel<<<(D_DIM + 255) / 256, 256, 0, stream>>>(W, dabs);

  // 2) Streaming scale: one block per 32 rows, 256 threads (8 wave32s).
  const int grid = (B + ROWS_PER_BLOCK - 1) / ROWS_PER_BLOCK;
  scale_kernel<<<grid, 256, 0, stream>>>(x, dabs, out, B);
}